// DoubleConv_5720896438572
// MI455X (gfx1250) — compile-verified
//
#include <hip/hip_runtime.h>

typedef __attribute__((ext_vector_type(2))) float v2f;
typedef __attribute__((ext_vector_type(8))) float v8f;

#define EPSV 1e-5f

// ---------------------------------------------------------------------------
// Sparse conv: out[i,:] = sum_k gather(feats, nbr[i,k]) @ W[k]
// One wave = 16 output voxels x 32 output channels (two 16x16 f32 WMMA tiles).
// Weights are packed into LDS in the exact B-matrix (4x16, f32) lane layout.
// FUSE_BN applies y = relu(x*scale[c]+shift[c]) to gathered values (conv2).
// ---------------------------------------------------------------------------
template <int CIN, bool FUSE_BN>
__global__ __launch_bounds__(256) void sparse_conv_wmma(
    const float* __restrict__ feats,   // [n, CIN]
    const int*   __restrict__ nbr,     // [n, 27]
    const float* __restrict__ W,       // [27, CIN, 32]
    const float* __restrict__ bnscale, // [32] or null
    const float* __restrict__ bnshift, // [32] or null
    float*       __restrict__ out,     // [n, 32]
    int n)
{
    constexpr int KS = CIN / 4;               // K-steps of 4 per offset
    extern __shared__ float lds_w[];          // 27*KS*2*64 floats

    const int tid  = threadIdx.x;
    const int lane = tid & 31;
    const int half = lane >> 4;               // 0: K pair {0,1}, 1: K pair {2,3}
    const int r    = lane & 15;               // row (A) / col (B,C,D) index

    // ---- pack W into LDS in B-matrix VGPR layout --------------------------
    // flat pair index i = ((k*KS + ks)*2 + nt)*32 + l ; element j in {0,1}
    // value = W[k][ks*4 + 2*(l>>4) + j][nt*16 + (l&15)]
    for (int i = tid; i < 27 * KS * 2 * 32; i += 256) {
        int l  = i & 31;
        int nt = (i >> 5) & 1;
        int t  = i >> 6;
        int ks = t % KS;
        int k  = t / KS;
        int c   = ks * 4 + 2 * (l >> 4);
        int col = nt * 16 + (l & 15);
        v2f wv;
        wv.x = W[(k * CIN + c) * 32 + col];
        wv.y = W[(k * CIN + c + 1) * 32 + col];
        *(v2f*)&lds_w[(size_t)i * 2] = wv;
    }
    __syncthreads();

    // ---- hoist BN params: channel pair per K-step is fixed per lane -------
    v2f sc[KS], sh[KS];
    if constexpr (FUSE_BN) {
#pragma unroll
        for (int ks = 0; ks < KS; ++ks) {
            int c = ks * 4 + 2 * half;
            sc[ks].x = bnscale[c];     sc[ks].y = bnscale[c + 1];
            sh[ks].x = bnshift[c];     sh[ks].y = bnshift[c + 1];
        }
    }

    const int wave = tid >> 5;
    const long long base = (long long)blockIdx.x * 128 + wave * 16;
    if (base >= n) return;                    // wave-uniform exit

    const long long row = base + r;           // voxel this lane gathers for
    const bool rowOK = row < (long long)n;

    v8f acc0 = {};                            // cout 0..15
    v8f acc1 = {};                            // cout 16..31

    for (int k = 0; k < 27; ++k) {
        int idx = -1;
        if (rowOK) idx = nbr[row * 27 + k];
        const bool ok = idx >= 0;
#pragma unroll
        for (int ks = 0; ks < KS; ++ks) {
            v2f a = {};
            if (ok) {
                a = *(const v2f*)(feats + (size_t)idx * CIN + ks * 4 + 2 * half);
                if constexpr (FUSE_BN) {
                    a.x = fmaxf(fmaf(a.x, sc[ks].x, sh[ks].x), 0.0f);
                    a.y = fmaxf(fmaf(a.y, sc[ks].y, sh[ks].y), 0.0f);
                }
            }
            const size_t wb = (size_t)(((k * KS + ks) * 2) * 32 + lane) * 2;
            v2f b0 = *(const v2f*)&lds_w[wb];
            v2f b1 = *(const v2f*)&lds_w[wb + 64 * 2];
            acc0 = __builtin_amdgcn_wmma_f32_16x16x4_f32(
                false, a, false, b0, (short)0, acc0, false, false);
            acc1 = __builtin_amdgcn_wmma_f32_16x16x4_f32(
                false, a, false, b1, (short)0, acc1, false, false);
        }
    }

    // ---- store D tiles: lane(r,half) holds col N=r, rows M = v + 8*half ---
#pragma unroll
    for (int v = 0; v < 8; ++v) {
        long long m = base + v + 8 * half;
        if (m < (long long)n) {
            out[m * 32 + r]      = acc0[v];
            out[m * 32 + 16 + r] = acc1[v];
        }
    }
}

// ---------------------------------------------------------------------------
// Per-channel sum / sum-of-squares over [n, 32] rows (for BN stats).
// ---------------------------------------------------------------------------
__global__ __launch_bounds__(256) void reduce_stats_kernel(
    const float* __restrict__ x, int n,
    float* __restrict__ sum, float* __restrict__ sumsq)
{
    __shared__ float ls[256], lq[256];
    const int c = threadIdx.x & 31;
    const int g = threadIdx.x >> 5;   // 0..7 row groups per block
    float s = 0.0f, q = 0.0f;
    for (long long i = (long long)blockIdx.x * 8 + g; i < n;
         i += (long long)gridDim.x * 8) {
        float v = x[i * 32 + c];
        s += v;
        q += v * v;
    }
    ls[threadIdx.x] = s;
    lq[threadIdx.x] = q;
    __syncthreads();
    if (threadIdx.x < 32) {
        float ts = 0.0f, tq = 0.0f;
        for (int w = 0; w < 8; ++w) {
            ts += ls[w * 32 + threadIdx.x];
            tq += lq[w * 32 + threadIdx.x];
        }
        atomicAdd(&sum[threadIdx.x], ts);
        atomicAdd(&sumsq[threadIdx.x], tq);
    }
}

__global__ void zero_stats_kernel(float* p) { p[threadIdx.x] = 0.0f; }

__global__ void finalize_bn_kernel(const float* __restrict__ sum,
                                   const float* __restrict__ sumsq,
                                   const float* __restrict__ gamma,
                                   const float* __restrict__ beta,
                                   float* __restrict__ scale,
                                   float* __restrict__ shift, float inv_n)
{
    int c = threadIdx.x;  // 32 threads
    float m   = sum[c] * inv_n;
    float var = sumsq[c] * inv_n - m * m;
    float is  = __frsqrt_rn(var + EPSV) * gamma[c];
    scale[c] = is;
    shift[c] = beta[c] - m * is;
}

__global__ __launch_bounds__(256) void apply_bn_kernel(
    float* __restrict__ x, long long total,
    const float* __restrict__ scale, const float* __restrict__ shift)
{
    long long i = (long long)blockIdx.x * 256 + threadIdx.x;
    const long long stride = (long long)gridDim.x * 256;
    for (; i < total; i += stride) {
        int c = (int)(i & 31);
        x[i] = fmaxf(fmaf(x[i], scale[c], shift[c]), 0.0f);
    }
}

// ---------------------------------------------------------------------------
extern "C" void kernel_launch(void* const* d_in, const int* in_sizes, int n_in,
                              void* d_out, int out_size, void* d_ws, size_t ws_size,
                              hipStream_t stream)
{
    const float* feats  = (const float*)d_in[0];
    const int*   nbr    = (const int*)  d_in[1];
    const float* W1     = (const float*)d_in[2];
    const float* gamma1 = (const float*)d_in[3];
    const float* beta1  = (const float*)d_in[4];
    const float* W2     = (const float*)d_in[5];
    const float* gamma2 = (const float*)d_in[6];
    const float* beta2  = (const float*)d_in[7];
    float* out = (float*)d_out;

    const int n = in_sizes[0] / 16;

    // workspace layout: h_raw [n*32] | sum1,sq1,sum2,sq2,scale1,shift1,scale2,shift2
    float* h_raw  = (float*)d_ws;
    float* stats  = h_raw + (size_t)n * 32;
    float* sum1   = stats;
    float* sq1    = stats + 32;
    float* sum2   = stats + 64;
    float* sq2    = stats + 96;
    float* scale1 = stats + 128;
    float* shift1 = stats + 160;
    float* scale2 = stats + 192;
    float* shift2 = stats + 224;

    const int convBlocks = (n + 127) / 128;
    const size_t lds1 = (size_t)27 * (16 / 4) * 2 * 64 * sizeof(float); //  55 KB
    const size_t lds2 = (size_t)27 * (32 / 4) * 2 * 64 * sizeof(float); // 110 KB
    const float inv_n = 1.0f / (float)n;

    zero_stats_kernel<<<1, 128, 0, stream>>>(stats);

    // conv1: feats[16] -> h_raw[32]
    sparse_conv_wmma<16, false><<<convBlocks, 256, lds1, stream>>>(
        feats, nbr, W1, nullptr, nullptr, h_raw, n);

    reduce_stats_kernel<<<1024, 256, 0, stream>>>(h_raw, n, sum1, sq1);
    finalize_bn_kernel<<<1, 32, 0, stream>>>(sum1, sq1, gamma1, beta1,
                                             scale1, shift1, inv_n);

    // conv2 with BN1+ReLU fused into the gather: h_raw[32] -> out[32]
    sparse_conv_wmma<32, true><<<convBlocks, 256, lds2, stream>>>(
        h_raw, nbr, W2, scale1, shift1, out, n);

    reduce_stats_kernel<<<1024, 256, 0, stream>>>(out, n, sum2, sq2);
    finalize_bn_kernel<<<1, 32, 0, stream>>>(sum2, sq2, gamma2, beta2,
                                             scale2, shift2, inv_n);

    apply_bn_kernel<<<2048, 256, 0, stream>>>(out, (long long)n * 32,
                                              scale2, shift2);
}